// Texture_73873437491454
// MI455X (gfx1250) — compile-verified
//
#include <hip/hip_runtime.h>
#include <hip/hip_bf16.h>

// Texture bilinear gather for MI455X (gfx1250).
// Memory-bound gather kernel: no matrix contraction exists (the reduction
// index couples to BOTH free indices), so WMMA/TDM are inapplicable by
// construction. Optimization levers: TH cache-policy (keep 256MB atlas hot in
// the 192MB L2 by streaming uv/mask/out non-temporally), b64 paired-x gathers,
// wave32-friendly 256-thread blocks, full channel unroll for load clustering.

namespace {

constexpr int kN      = 4;
constexpr int kLayers = 4;
constexpr int kH      = 512;
constexpr int kW      = 512;
constexpr int kHW     = kH * kW;
constexpr int kT      = 64;
constexpr int kFc     = 16;
constexpr int kD      = 256;
constexpr int kDD     = kD * kD;

// 8-byte pair at 4-byte alignment: lets the backend emit global_load_b64 even
// when x0 is odd (gfx1250 supports dword-aligned b64 loads).
struct __attribute__((packed, aligned(4))) F2 { float x, y; };

__global__ __launch_bounds__(256) void texture_sample_kernel(
    const float* __restrict__ uv,    // (N, 2*L, H, W)
    const int*   __restrict__ mask,  // (N, L, H, W)
    const float* __restrict__ data,  // (T, Fc, D, D)
    float*       __restrict__ out)   // (N, L*Fc, H, W)
{
  const int p     = blockIdx.x * 256 + threadIdx.x;  // pixel in H*W
  const int layer = blockIdx.y;
  const int n     = blockIdx.z;

  // Streamed-once inputs: non-temporal so they don't evict the atlas from L2.
  const size_t uvOff = ((size_t)n * (2 * kLayers) + 2 * layer) * kHW + p;
  const float ux  = __builtin_nontemporal_load(uv + uvOff);
  const float uy  = __builtin_nontemporal_load(uv + uvOff + kHW);
  const int   tid = __builtin_nontemporal_load(
      mask + ((size_t)n * kLayers + layer) * kHW + p);

  float* outp = out + ((size_t)n * (kLayers * kFc) + layer * kFc) * kHW + p;

  if (tid < 1) {  // masked-out pixel: zero all 16 channels, skip all gathers
#pragma unroll
    for (int c = 0; c < kFc; ++c)
      __builtin_nontemporal_store(0.0f, outp + (size_t)c * kHW);
    return;
  }

  const int st = tid < (kT - 1) ? tid : (kT - 1);  // clamp(tid, 0, T-1)

  // Mirror reference op order exactly: ((u+1)*D - 1) * 0.5, then clamp.
  float ix = ((ux + 1.0f) * (float)kD - 1.0f) * 0.5f;
  ix = fminf(fmaxf(ix, 0.0f), (float)(kD - 1));
  float iy = ((uy + 1.0f) * (float)kD - 1.0f) * 0.5f;
  iy = fminf(fmaxf(iy, 0.0f), (float)(kD - 1));

  const float x0f = floorf(ix);
  const float y0f = floorf(iy);
  float       wx  = ix - x0f;
  const float wy  = iy - y0f;
  int       x0 = (int)x0f;
  const int y0 = (int)y0f;

  // Fold the x-edge clamp into the b64 window: if x0==D-1 then ix==D-1
  // exactly (clamped) and original wx==0; shifting to x0=D-2 with wx=1.0
  // selects the same texel. All other cases unchanged.
  if (x0 > kD - 2) { x0 = kD - 2; wx = ix - (float)x0; }
  int y1 = y0 + 1; if (y1 > kD - 1) y1 = kD - 1;

  const float w00 = (1.0f - wx) * (1.0f - wy);
  const float w01 = wx * (1.0f - wy);
  const float w10 = (1.0f - wx) * wy;
  const float w11 = wx * wy;

  // Two row base pointers; all 16 channel offsets (c * 256 KB, max 3.93 MB)
  // fit in the signed 24-bit instruction offset.
  const float* base = data + (size_t)st * (kFc * kDD) + x0;
  const F2* row0 = (const F2*)(base + y0 * kD);
  const F2* row1 = (const F2*)(base + y1 * kD);

#pragma unroll
  for (int c = 0; c < kFc; ++c) {
    const F2 g0 = row0[(size_t)c * (kDD / 2)];  // [x0, x0+1] @ row y0
    const F2 g1 = row1[(size_t)c * (kDD / 2)];  // [x0, x0+1] @ row y1
    const float v = g0.x * w00 + g0.y * w01 + g1.x * w10 + g1.y * w11;
    __builtin_nontemporal_store(v, outp + (size_t)c * kHW);  // write-once
  }
}

}  // namespace

extern "C" void kernel_launch(void* const* d_in, const int* in_sizes, int n_in,
                              void* d_out, int out_size, void* d_ws, size_t ws_size,
                              hipStream_t stream) {
  // setup_inputs order: uv_inputs, mask_inputs, world_positions(unused),
  // extrinsics(unused), data
  const float* uv   = (const float*)d_in[0];
  const int*   mask = (const int*)d_in[1];
  const float* data = (const float*)d_in[4];
  float*       out  = (float*)d_out;

  dim3 grid(kHW / 256, kLayers, kN);
  texture_sample_kernel<<<grid, 256, 0, stream>>>(uv, mask, data, out);
}